// FeatureExpander_49185965473877
// MI455X (gfx1250) — compile-verified
//
#include <hip/hip_runtime.h>
#include <hip/hip_bf16.h>

typedef __attribute__((ext_vector_type(2))) float v2f;
typedef __attribute__((ext_vector_type(8))) float v8f;

// Problem constants (match reference)
#define PP 2000
#define DD 512
#define BB 64

// One wave32 computes a 16(batch) x 16(d) tile of out[:, p, :] for one p via
// V_WMMA_F32_16X16X4_F32:
//   A (16x4): col K=0 = x[btile*16 + m, p], K=1..3 zero
//   B (4x16): row K=0 = W[p, ntile*16 + n], K=1..3 zero
//   C (16x16): C[m,n] = b[p, ntile*16 + n]  (broadcast over m)
//   D = A*B + C = x*W + b  exactly.
__global__ __launch_bounds__(256) void
FeatureExpander_wmma_kernel(const float* __restrict__ x,
                            const float* __restrict__ W,
                            const float* __restrict__ bias,
                            float* __restrict__ out)
{
    const int NT = DD / 16;  // 32 d-tiles
    const int BT = BB / 16;  // 4 batch-tiles

    const int gtid = blockIdx.x * blockDim.x + threadIdx.x;
    const int wave = gtid >> 5;          // global wave id
    const int lane = threadIdx.x & 31;

    // ntile fastest -> consecutive waves write consecutive 64B segments
    const int ntile = wave % NT;
    const int rest  = wave / NT;
    const int p     = rest % PP;
    const int btile = rest / PP;
    if (btile >= BT) return;             // wave-uniform guard (EXEC stays all-1)

    const int  l16 = lane & 15;
    const bool hi  = (lane >= 16);

    // Gather the three broadcast operands (all hit L2 after first touch).
    const float xv = x[(size_t)(btile * 16 + l16) * PP + p];
    const float wv = W[(size_t)p * DD + ntile * 16 + l16];
    const float bv = bias[(size_t)p * DD + ntile * 16 + l16];

    // A: VGPR0 lanes 0-15 hold K=0 column; everything else zero.
    v2f a;  a.x = hi ? 0.0f : xv;  a.y = 0.0f;
    // B: VGPR0 lanes 0-15 hold K=0 row; everything else zero.
    v2f bm; bm.x = hi ? 0.0f : wv; bm.y = 0.0f;
    // C: broadcast b[p, n] across all 8 M-rows held per lane.
    v8f c = {bv, bv, bv, bv, bv, bv, bv, bv};

    // (neg_a, A, neg_b, B, c_mod, C, reuse_a, reuse_b)
    v8f d = __builtin_amdgcn_wmma_f32_16x16x4_f32(
        false, a, false, bm, (short)0, c, false, false);

    // D layout: VGPR r -> M = r + (hi ? 8 : 0), N = lane % 16.
    const size_t row_stride = (size_t)PP * DD;                    // between batch rows
    size_t base = (size_t)(btile * 16 + (hi ? 8 : 0)) * row_stride
                + (size_t)p * DD + (size_t)(ntile * 16 + l16);

#pragma unroll
    for (int r = 0; r < 8; ++r) {
        // 262 MB streaming write-once output: non-temporal so it does not
        // evict the L2-resident W/b working set (8 MB, reused 64x).
        __builtin_nontemporal_store(d[r], out + base + (size_t)r * row_stride);
    }
}

extern "C" void kernel_launch(void* const* d_in, const int* in_sizes, int n_in,
                              void* d_out, int out_size, void* d_ws, size_t ws_size,
                              hipStream_t stream)
{
    (void)in_sizes; (void)n_in; (void)out_size; (void)d_ws; (void)ws_size;

    const float* x    = (const float*)d_in[0];
    const float* W    = (const float*)d_in[1];
    const float* bias = (const float*)d_in[2];
    float*       out  = (float*)d_out;

    const int waves   = (BB / 16) * PP * (DD / 16);   // 256,000 waves
    const int threads = 256;                          // 8 waves / block
    const int blocks  = (waves * 32 + threads - 1) / threads;  // 32,000

    FeatureExpander_wmma_kernel<<<blocks, threads, 0, stream>>>(x, W, bias, out);
}